// WaveletTokenizer_79353815761238
// MI455X (gfx1250) — compile-verified
//
#include <hip/hip_runtime.h>
#include <math.h>

typedef float v2f __attribute__((ext_vector_type(2)));
typedef float v8f __attribute__((ext_vector_type(8)));

#define VOCAB    4096
#define L_PER_B  97536
#define N_TOK    195072        // B=2 * L_PER_B
#define G1 73728
#define G2 18432
#define G3 4608
#define G4 768
#define OFF2 73728
#define OFF3 92160
#define OFF4 96768
#define NBLK_VQ 762            // N_TOK / (8 waves * 32 tokens)

__device__ __forceinline__ float group_weight(int t_local) {
    // per-group 1/(B * L_g * 4), B = 2
    if (t_local < OFF2) return 1.0f / (2.0f * (float)G1 * 4.0f);
    if (t_local < OFF3) return 1.0f / (2.0f * (float)G2 * 4.0f);
    if (t_local < OFF4) return 1.0f / (2.0f * (float)G3 * 4.0f);
    return 1.0f / (2.0f * (float)G4 * 4.0f);
}

// ---------------------------------------------------------------------------
// Kernel 1: build quaternion tokens (N_TOK x 4 floats) in workspace, in the
// exact concatenated (b, [g1 g2 g3 g4]) order of the reference.
// ---------------------------------------------------------------------------
__global__ void wt_prep_feats(const float* __restrict__ yh1,
                              const float* __restrict__ yh2,
                              const float* __restrict__ yh3,
                              const float* __restrict__ yl,
                              float* __restrict__ feats) {
    int T = blockIdx.x * blockDim.x + threadIdx.x;
    if (T >= N_TOK) return;
    int b = T / L_PER_B;
    int t = T - b * L_PER_B;
    float4 q;
    if (t < OFF4) {
        const float* src; int H; int tl;
        if (t < OFF2)      { src = yh1; H = 64; tl = t; }
        else if (t < OFF3) { src = yh2; H = 32; tl = t - OFF2; }
        else               { src = yh3; H = 16; tl = t - OFF3; }
        // t_local = ((h*W + w)*6 + o)*3 + c   with W == H
        int c = tl % 3;
        int o = (tl / 3) % 6;
        int w = (tl / 18) % H;
        int h = tl / (18 * H);
        int base = ((((b * 3 + c) * 6 + o) * H + h) * H + w) * 2;
        float re = src[base];
        float im = src[base + 1];
        float amp = sqrtf(re * re + im * im);
        float ph  = atan2f(im, re);
        q = make_float4(re, im, amp, ph);
    } else {
        int tl = t - OFF4;
        int c = tl % 3;
        int w = (tl / 3) % 16;
        int h = tl / 48;
        float v = yl[((b * 3 + c) * 16 + h) * 16 + w];
        q = make_float4(v, 0.0f, fabsf(v), 0.0f);
    }
    ((float4*)feats)[T] = q;
}

// branch-free (dist, idx) combine against a ds_swizzle'd partner lane
#define REDUCE_STEP(MASK)                                                      \
    _Pragma("unroll")                                                          \
    for (int i = 0; i < 16; ++i) {                                             \
        float od = __int_as_float(__builtin_amdgcn_ds_swizzle(                 \
            __float_as_int(best[i]), ((MASK) << 10) | 0x1F));                  \
        int oi = __builtin_amdgcn_ds_swizzle(bidx[i], ((MASK) << 10) | 0x1F);  \
        bool take = (od < best[i]) | ((od == best[i]) & (oi < bidx[i]));       \
        best[i] = take ? od : best[i];                                         \
        bidx[i] = take ? oi : bidx[i];                                         \
    }

// ---------------------------------------------------------------------------
// Kernel 2: WMMA-based VQ.  One wave handles 32 tokens (two A matrices) vs
// all 4096 codes: per 16-code tile, one B load feeds two
// V_WMMA_F32_16X16X4_F32 ops:  D = (-2*X) x E^T + ||e||^2.
// Software-pipelined with prefetch distance 1.
// ---------------------------------------------------------------------------
__global__ __launch_bounds__(256) void wt_vq_wmma(
        const float* __restrict__ feats,
        const float* __restrict__ cb,
        float* __restrict__ out_quant,    // N_TOK x 4
        float* __restrict__ out_idx,      // N_TOK (as float)
        float* __restrict__ partials) {   // one loss partial per block
    __shared__ float e2[VOCAB];           // 16 KB
    __shared__ float wls[16];

    int tid = threadIdx.x;
    for (int n = tid; n < VOCAB; n += 256) {
        float4 e = ((const float4*)cb)[n];
        e2[n] = e.x * e.x + e.y * e.y + e.z * e.z + e.w * e.w;
    }
    __syncthreads();

    int lane   = tid & 31;
    int waveId = tid >> 5;
    int token_base = (blockIdx.x * 8 + waveId) * 32;

    int mrow  = lane & 15;      // matrix row (A) / column (B,C) served by lane
    int khalf = lane >> 4;      // 0: K=0,1   1: K=2,3

    // Two A matrices (16x4 f32 each), pre-scaled by -2:
    //   VGPR0 = K(2*khalf), VGPR1 = K(2*khalf+1)
    const float* fp0 = feats + (size_t)(token_base + mrow) * 4 + khalf * 2;
    const float* fp1 = fp0 + 16 * 4;
    v2f a0, a1;
    a0.x = -2.0f * fp0[0];  a0.y = -2.0f * fp0[1];
    a1.x = -2.0f * fp1[0];  a1.y = -2.0f * fp1[1];

    float best[16];
    int   bidx[16];
#pragma unroll
    for (int i = 0; i < 16; ++i) { best[i] = 3.4e38f; bidx[i] = 0; }

    // prologue: load tile 0's B column slice and bias
    int n0 = mrow;
    v2f   bm;
    bm.x = cb[(size_t)n0 * 4 + khalf * 2];
    bm.y = cb[(size_t)n0 * 4 + khalf * 2 + 1];
    float cv = e2[n0];

#pragma unroll 2
    for (int tile = 0; tile < 256; ++tile) {
        // prefetch next tile (wraps to tile 0 on the tail, branch-free & in-bounds)
        int nn = (((tile + 1) & 255) << 4) + mrow;
        const float* bp = cb + (size_t)nn * 4 + khalf * 2;
        v2f bm_n;
        bm_n.x = bp[0];
        bm_n.y = bp[1];
        float cv_n = e2[nn];

        v8f c0 = { cv, cv, cv, cv, cv, cv, cv, cv };
        v8f d0 = __builtin_amdgcn_wmma_f32_16x16x4_f32(
            false, a0, false, bm, (short)0, c0, false, false);
        v8f d1 = __builtin_amdgcn_wmma_f32_16x16x4_f32(
            false, a1, false, bm, (short)0, c0, false, false);

        int n = (tile << 4) + mrow;
#pragma unroll
        for (int i = 0; i < 8; ++i) {
            bool lt0 = d0[i] < best[i];
            best[i]     = lt0 ? d0[i] : best[i];
            bidx[i]     = lt0 ? n     : bidx[i];
            bool lt1 = d1[i] < best[i + 8];
            best[i + 8] = lt1 ? d1[i] : best[i + 8];
            bidx[i + 8] = lt1 ? n     : bidx[i + 8];
        }
        bm = bm_n;
        cv = cv_n;
    }

    // argmin across the 16 columns (xor swizzles stay inside each 16-lane half)
    REDUCE_STEP(8)
    REDUCE_STEP(4)
    REDUCE_STEP(2)
    REDUCE_STEP(1)

    // lanes 0 and 16 hold results: slot i<8 -> matrix0 token khalf*8+i,
    // slot i>=8 -> matrix1 token 16 + khalf*8 + (i-8)
    if (mrow == 0) {
        float lloss = 0.0f;
#pragma unroll
        for (int i = 0; i < 16; ++i) {
            int t = token_base + (i < 8 ? (khalf * 8 + i)
                                        : (16 + khalf * 8 + (i - 8)));
            int id = bidx[i];
            float4 e = ((const float4*)cb)[id];
            float4 f = ((const float4*)feats)[t];
            ((float4*)out_quant)[t] = e;      // straight-through == quant
            out_idx[t] = (float)id;
            float w  = group_weight(t % L_PER_B);
            float dx = e.x - f.x, dy = e.y - f.y;
            float dz = e.z - f.z, dw = e.w - f.w;
            lloss += w * (dx * dx + dy * dy + dz * dz + dw * dw);
        }
        wls[waveId * 2 + khalf] = lloss;
    }
    __syncthreads();
    if (tid == 0) {
        float s = 0.0f;
        for (int i = 0; i < 16; ++i) s += wls[i];
        partials[blockIdx.x] = s;
    }
}

// ---------------------------------------------------------------------------
// Kernel 3: deterministic single-block reduction of loss partials.
// ---------------------------------------------------------------------------
__global__ void wt_reduce_loss(const float* __restrict__ partials, int n,
                               float* __restrict__ loss_out) {
    __shared__ float s[256];
    float acc = 0.0f;
    for (int i = threadIdx.x; i < n; i += 256) acc += partials[i];
    s[threadIdx.x] = acc;
    __syncthreads();
    for (int m = 128; m >= 1; m >>= 1) {
        if (threadIdx.x < m) s[threadIdx.x] += s[threadIdx.x + m];
        __syncthreads();
    }
    if (threadIdx.x == 0) *loss_out = s[0];
}

extern "C" void kernel_launch(void* const* d_in, const int* in_sizes, int n_in,
                              void* d_out, int out_size, void* d_ws, size_t ws_size,
                              hipStream_t stream) {
    (void)in_sizes; (void)n_in; (void)out_size; (void)ws_size;
    const float* yh1 = (const float*)d_in[0];
    const float* yh2 = (const float*)d_in[1];
    const float* yh3 = (const float*)d_in[2];
    const float* yl  = (const float*)d_in[3];
    const float* cb  = (const float*)d_in[4];

    float* feats    = (float*)d_ws;                 // N_TOK*4 floats (~3 MB)
    float* partials = feats + (size_t)N_TOK * 4;    // NBLK_VQ floats

    float* out_quant = (float*)d_out;                       // 780288 floats
    float* out_idx   = out_quant + (size_t)N_TOK * 4;       // 195072 floats
    float* out_loss  = out_idx + N_TOK;                     // 1 float

    wt_prep_feats<<<(N_TOK + 255) / 256, 256, 0, stream>>>(yh1, yh2, yh3, yl, feats);
    wt_vq_wmma<<<NBLK_VQ, 256, 0, stream>>>(feats, cb, out_quant, out_idx, partials);
    wt_reduce_loss<<<1, 256, 0, stream>>>(partials, NBLK_VQ, out_loss);
}